// Classifier_F_87917980549692
// MI455X (gfx1250) — compile-verified
//
#include <hip/hip_runtime.h>

typedef float v2f __attribute__((ext_vector_type(2)));
typedef float v8f __attribute__((ext_vector_type(8)));

#define WAVES_PER_BLOCK 8

// Out[M x NT*16] = X[M x K] @ W[NT*16 x K]^T (+ bias).
// One wave32 owns a 16-row stripe: preloads its A fragment (all K) once into
// registers, then for each 16-col tile preloads B as one clause and runs a
// back-to-back V_WMMA_F32_16X16X4_F32 chain (accumulator-chained, hazard-free).
// MEANPAIR: X_eff[n][k] = 0.5*(X[n*2K + k] + X[n*2K + K + k])  (fused emb-mean)
template <int K, int NT, bool MEANPAIR>
__global__ __launch_bounds__(32 * WAVES_PER_BLOCK)
void wmma_gemm_kernel(const float* __restrict__ X, const float* __restrict__ W,
                      const float* __restrict__ bias, float* __restrict__ Out, int M)
{
    constexpr int KS = K / 4;      // WMMA K-steps
    constexpr int NO = NT * 16;    // output width
    const int lane = threadIdx.x & 31;
    const int wave = blockIdx.x * (blockDim.x >> 5) + (threadIdx.x >> 5);
    if (wave >= (M >> 4)) return;            // wave-uniform: EXEC stays all-1s
    const int half = lane >> 4;              // K pair {0,1} vs {2,3} within step
    const int lm   = lane & 15;
    const int arow = wave * 16 + lm;         // A: lane holds row arow

    // ---- preload A fragment for the whole K extent (issued as load clauses) ----
    v2f a[KS];
    if (MEANPAIR) {
        const float* __restrict__ x0 = X + (size_t)arow * 2 * K;
#pragma unroll
        for (int i = 0; i < KS; ++i) {
            const int ka = 4 * i + half * 2;
            const v2f lo = *(const v2f*)(x0 + ka);
            const v2f hi = *(const v2f*)(x0 + K + ka);
            a[i].x = 0.5f * (lo.x + hi.x);
            a[i].y = 0.5f * (lo.y + hi.y);
        }
    } else {
        const float* __restrict__ x0 = X + (size_t)arow * K;
#pragma unroll
        for (int i = 0; i < KS; ++i)
            a[i] = *(const v2f*)(x0 + 4 * i + half * 2);
    }

    // ---- for each 16-wide output tile: clause-load B, then WMMA chain ----
#pragma unroll
    for (int nt = 0; nt < NT; ++nt) {
        const int bcol = nt * 16 + lm;       // B col == W row
        const float* __restrict__ wrow = W + (size_t)bcol * K;
        v2f b[KS];
#pragma unroll
        for (int i = 0; i < KS; ++i)
            b[i] = *(const v2f*)(wrow + 4 * i + half * 2);

        v8f acc = {0.f, 0.f, 0.f, 0.f, 0.f, 0.f, 0.f, 0.f};
#pragma unroll
        for (int i = 0; i < KS; ++i)
            acc = __builtin_amdgcn_wmma_f32_16x16x4_f32(false, a[i], false, b[i],
                                                        (short)0, acc, false, false);

        // C/D layout: VGPR r, lanes 0-15 -> D[M=r][N=lane]; lanes 16-31 -> D[M=8+r]
        const int   ocol = nt * 16 + lm;
        const float bv   = bias ? bias[ocol] : 0.0f;
#pragma unroll
        for (int r = 0; r < 8; ++r) {
            const int orow = wave * 16 + half * 8 + r;
            Out[(size_t)orow * NO + ocol] = acc[r] + bv;
        }
    }
}

__global__ void deg_kernel(const int* __restrict__ dst, float* __restrict__ deg, int E)
{
    int i = blockIdx.x * blockDim.x + threadIdx.x;
    if (i < E) atomicAdd(&deg[dst[i]], 1.0f);
}

// Per-edge scatter of C floats (C = 1<<lc2 float4 chunks per edge).
__global__ void scatter_kernel(const float* __restrict__ P, const int* __restrict__ src,
                               const int* __restrict__ dst, float* __restrict__ Agg,
                               int E, int C, int lc2)
{
    long long tid   = (long long)blockIdx.x * blockDim.x + threadIdx.x;
    long long total = (long long)E << lc2;
    if (tid >= total) return;
    const int e = (int)(tid >> lc2);
    const int c = ((int)tid & ((1 << lc2) - 1)) << 2;
    const int s = src[e];
    const int d = dst[e];
    const float4 v = *(const float4*)(P + (size_t)s * C + c);
    float* ap = Agg + (size_t)d * C + c;
    atomicAdd(ap + 0, v.x);
    atomicAdd(ap + 1, v.y);
    atomicAdd(ap + 2, v.z);
    atomicAdd(ap + 3, v.w);
}

// h = relu(agg/max(deg,1) + bl[c] + r + e0), C = 64
__global__ void combine_h_kernel(const float* __restrict__ agg, const float* __restrict__ r0,
                                 const float* __restrict__ e0, const float* __restrict__ bl,
                                 const float* __restrict__ deg, float* __restrict__ h, int N)
{
    long long t = (long long)blockIdx.x * blockDim.x + threadIdx.x;
    if (t >= (long long)N * 64) return;
    const int n = (int)(t >> 6);
    const int c = (int)t & 63;
    float dv = deg[n];
    dv = dv > 1.0f ? dv : 1.0f;
    const float v = agg[t] / dv + bl[c] + r0[t] + e0[t];
    h[t] = v > 0.0f ? v : 0.0f;
}

// out = agg1/max(deg,1) + bl1[c] + r1 + e1, C = 16
__global__ void final_kernel(const float* __restrict__ agg1, const float* __restrict__ r1,
                             const float* __restrict__ e1, const float* __restrict__ bl1,
                             const float* __restrict__ deg, float* __restrict__ out, int N)
{
    int t = blockIdx.x * blockDim.x + threadIdx.x;
    if (t >= N * 16) return;
    const int n = t >> 4;
    const int c = t & 15;
    float dv = deg[n];
    dv = dv > 1.0f ? dv : 1.0f;
    out[t] = agg1[t] / dv + bl1[c] + r1[t] + e1[t];
}

extern "C" void kernel_launch(void* const* d_in, const int* in_sizes, int n_in,
                              void* d_out, int out_size, void* d_ws, size_t ws_size,
                              hipStream_t stream)
{
    const float* x_feat = (const float*)d_in[0];
    const float* x_emb  = (const float*)d_in[1];
    const int*   eidx   = (const int*)d_in[2];
    const float* Wl0 = (const float*)d_in[3];
    const float* bl0 = (const float*)d_in[4];
    const float* Wr0 = (const float*)d_in[5];
    const float* We0 = (const float*)d_in[6];
    const float* be0 = (const float*)d_in[7];
    const float* Wl1 = (const float*)d_in[8];
    const float* bl1 = (const float*)d_in[9];
    const float* Wr1 = (const float*)d_in[10];
    const float* We1 = (const float*)d_in[11];
    const float* be1 = (const float*)d_in[12];

    const int FEAT = 128;
    const int N = in_sizes[0] / FEAT;
    const int E = in_sizes[2] / 2;
    const int* src = eidx;
    const int* dst = eidx + E;

    // Workspace layout (floats): total 257*N  (~103 MB for N=100k)
    float*       ws = (float*)d_ws;
    const size_t n  = (size_t)N;
    float* deg  = ws;                 // N
    float* e0   = ws + n;             // 64N
    float* p0   = ws + 65 * n;        // 64N (reused as h)
    float* r0   = ws + 129 * n;       // 64N
    float* agg0 = ws + 193 * n;       // 64N (region reused below after combine)
    float* h    = p0;
    float* p1   = agg0;               // 16N
    float* r1   = ws + 209 * n;       // 16N
    float* e1   = ws + 225 * n;       // 16N
    float* agg1 = ws + 241 * n;       // 16N, ends at 257N

    hipMemsetAsync(deg,  0, n * sizeof(float), stream);
    hipMemsetAsync(agg0, 0, 64 * n * sizeof(float), stream);

    deg_kernel<<<(E + 255) / 256, 256, 0, stream>>>(dst, deg, E);

    const int mwaves  = N >> 4;
    const int gblocks = (mwaves + WAVES_PER_BLOCK - 1) / WAVES_PER_BLOCK;
    const int gthreads = 32 * WAVES_PER_BLOCK;

    // Layer 0 projections (project first, aggregate in low-dim space)
    wmma_gemm_kernel<64, 4, true ><<<gblocks, gthreads, 0, stream>>>(x_emb,  We0, be0,     e0, N);
    wmma_gemm_kernel<128, 4, false><<<gblocks, gthreads, 0, stream>>>(x_feat, Wl0, nullptr, p0, N);
    wmma_gemm_kernel<128, 4, false><<<gblocks, gthreads, 0, stream>>>(x_feat, Wr0, nullptr, r0, N);

    {   // scatter p0[src] -> agg0[dst], C=64 (16 float4 chunks/edge)
        const long long total = (long long)E << 4;
        scatter_kernel<<<(int)((total + 255) / 256), 256, 0, stream>>>(p0, src, dst, agg0, E, 64, 4);
    }

    {   // h = relu(agg0/deg + bl0 + r0 + e0)  (overwrites p0)
        const long long total = (long long)N * 64;
        combine_h_kernel<<<(int)((total + 255) / 256), 256, 0, stream>>>(agg0, r0, e0, bl0, deg, h, N);
    }

    // Layer 1 projections (reuse agg0 region, free after combine)
    wmma_gemm_kernel<64, 1, false><<<gblocks, gthreads, 0, stream>>>(h,  Wl1, nullptr, p1, N);
    wmma_gemm_kernel<64, 1, false><<<gblocks, gthreads, 0, stream>>>(h,  Wr1, nullptr, r1, N);
    wmma_gemm_kernel<64, 1, false><<<gblocks, gthreads, 0, stream>>>(e0, We1, be1,     e1, N);

    hipMemsetAsync(agg1, 0, 16 * n * sizeof(float), stream);

    {   // scatter p1[src] -> agg1[dst], C=16 (4 float4 chunks/edge)
        const long long total = (long long)E << 2;
        scatter_kernel<<<(int)((total + 255) / 256), 256, 0, stream>>>(p1, src, dst, agg1, E, 16, 2);
    }

    final_kernel<<<(N * 16 + 255) / 256, 256, 0, stream>>>(agg1, r1, e1, bl1, deg,
                                                           (float*)d_out, N);
}